// ShiftVarConv2D_5866925326591
// MI455X (gfx1250) — compile-verified
//
#include <hip/hip_runtime.h>

// ShiftVarConv2D on MI455X (gfx1250): position-dependent 3x3 conv, 1->16 ch.
// out[n,m,Y,X] = bias[g*16+m] + sum_i W[(g*16+m)*9+i] * in[n,Y+i/3-1,X+i%3-1],
// g = (Y%8)*8 + X%8.  Per (row, X%8) tile: D[16 px x 16 ch] = A[16x12] * W^T[12x16]
// via 3x V_WMMA_F32_16X16X4_F32 (K=9 padded to 12), exact f32.
// Memory-bound (~140MB, ~0.6 GFLOP): async global->LDS input staging (halo),
// L2-resident weights, async LDS->global coalesced b128 output stores.

typedef float v2f __attribute__((ext_vector_type(2)));
typedef float v8f __attribute__((ext_vector_type(8)));
typedef int   v4i __attribute__((ext_vector_type(4)));

#define NB   128
#define HH   128
#define MCH  16
#define RWS  4               // output rows per workgroup
#define LDSW 136             // s_in row stride: [3]=left halo, [4..131]=payload(16B aligned), [132]=right halo
#define SLC  (RWS * HH + 4)  // 516: s_out per-channel stride (bank-conflict pad)

#if __has_builtin(__builtin_amdgcn_global_load_async_to_lds_b128) && \
    __has_builtin(__builtin_amdgcn_global_store_async_from_lds_b128)
#define USE_ASYNC_LDS 1
typedef __attribute__((address_space(1))) v4i* gptr_b128;   // global-side param type
typedef __attribute__((address_space(3))) v4i* lptr_b128;   // LDS-side param type
#else
#define USE_ASYNC_LDS 0
#endif

static __device__ __forceinline__ void wait_async_all() {
#if __has_builtin(__builtin_amdgcn_s_wait_asynccnt)
  __builtin_amdgcn_s_wait_asynccnt(0);
#else
  asm volatile("s_wait_asynccnt 0x0" ::: "memory");
#endif
}

__global__ __launch_bounds__(256) void shiftvar_wmma_kernel(
    const float* __restrict__ in,    // [128,1,128,128]
    const float* __restrict__ wgt,   // flat [(g*16+m)*9 + i]
    const float* __restrict__ bias,  // [1024]
    float* __restrict__ out)         // [128,16,128,128]
{
  __shared__ __align__(16) float s_in[(RWS + 2) * LDSW];
  __shared__ __align__(16) float s_out[MCH * SLC];

  const int tid  = threadIdx.x;
  const int lane = tid & 31;
  const int bx   = tid >> 5;            // wave id == X%8 group
  const int n     = blockIdx.x >> 5;    // image
  const int chunk = blockIdx.x & 31;    // 32 chunks of 4 rows
  const int Ybase = chunk * RWS;

  const float* inN = in + (size_t)n * (HH * HH);

  // ---- stage input rows Ybase-1 .. Ybase+RWS (zero halo) into LDS ----
#if USE_ASYNC_LDS
  if (tid < (RWS + 2) * (HH / 4)) {     // 192 lanes, one float4 each
    int r = tid >> 5, x4 = tid & 31;
    int gy = Ybase - 1 + r;
    float* ldst = &s_in[r * LDSW + 4 + x4 * 4];
    if (gy >= 0 && gy < HH) {
      __builtin_amdgcn_global_load_async_to_lds_b128(
          (gptr_b128)(inN + gy * HH + x4 * 4), (lptr_b128)ldst, 0, 0);
    } else {
      *(float4*)ldst = make_float4(0.f, 0.f, 0.f, 0.f);
    }
  }
#else
  for (int idx = tid; idx < (RWS + 2) * HH; idx += 256) {
    int r = idx >> 7, x = idx & 127;
    int gy = Ybase - 1 + r;
    s_in[r * LDSW + 4 + x] = (gy >= 0 && gy < HH) ? inN[gy * HH + x] : 0.0f;
  }
#endif
  if (tid < RWS + 2) {
    s_in[tid * LDSW + 3] = 0.0f;        // left halo (X = -1)
    s_in[tid * LDSW + 4 + HH] = 0.0f;   // right halo (X = 128)
  }
  // prefetch next chunk's input rows (gfx1250 global_prefetch_b8 path)
  if (Ybase + RWS + 5 < HH && tid < 48)
    __builtin_prefetch(inN + (Ybase + RWS - 1) * HH + tid * 16, 0, 1);

#if USE_ASYNC_LDS
  wait_async_all();
#endif
  __syncthreads();

  const int m  = lane & 15;        // pixel index p for A rows == channel for B/D cols
  const int hi = lane >> 4;
  const int kk = hi << 1;          // per-half K sub-offset (same pattern for A and B)

  for (int yloc = 0; yloc < RWS; ++yloc) {
    const int g = ((Ybase + yloc) & 7) * 8 + bx;   // (Y%8)*8 + X%8
    const float* wrow = wgt + (g * MCH + m) * 9;
    const float bv = bias[g * MCH + m];
    v8f c = {bv, bv, bv, bv, bv, bv, bv, bv};
#pragma unroll
    for (int j = 0; j < 3; ++j) {                  // K blocks {0..3},{4..7},{8..11}
      const int i0 = 4 * j + kk, i1 = i0 + 1;
      const int i0c = (i0 < 9) ? i0 : 8;           // clamp -> unconditional loads,
      const int i1c = (i1 < 9) ? i1 : 8;           // zero via cndmask (no exec branches)
      // A[p,i] = s_in[(yloc + i/3)*LDSW + 3 + bx + 8p + i%3]
      float ax = s_in[(yloc + i0c / 3) * LDSW + 3 + bx + 8 * m + i0c % 3];
      float ay = s_in[(yloc + i1c / 3) * LDSW + 3 + bx + 8 * m + i1c % 3];
      float wx = wrow[i0c];
      float wy = wrow[i1c];
      v2f a, b;
      a.x = (i0 < 9) ? ax : 0.0f;
      a.y = (i1 < 9) ? ay : 0.0f;
      b.x = (i0 < 9) ? wx : 0.0f;
      b.y = (i1 < 9) ? wy : 0.0f;
      c = __builtin_amdgcn_wmma_f32_16x16x4_f32(false, a, false, b, (short)0, c,
                                                false, false);
    }
    // D: VGPR v, lane l -> pixel p = v + 8*(l/16), channel m = l%16
#pragma unroll
    for (int v = 0; v < 8; ++v) {
      int pp = v + (hi << 3);
      s_out[m * SLC + yloc * HH + bx + 8 * pp] = c[v];
    }
  }
  __syncthreads();

  // ---- coalesced copy-out: s_out[m][yloc][X] -> out[n,m,Ybase+yloc,X] ----
  float* obase = out + (size_t)n * MCH * HH * HH + (size_t)Ybase * HH;
#if USE_ASYNC_LDS
  for (int idx = tid; idx < MCH * RWS * HH / 4; idx += 256) {
    int mm  = idx >> 7;        // / (RWS*HH/4 = 128)
    int rem = idx & 127;       // yloc*32 + X/4
    __builtin_amdgcn_global_store_async_from_lds_b128(
        (gptr_b128)(obase + (size_t)mm * (HH * HH) + rem * 4),
        (lptr_b128)&s_out[mm * SLC + rem * 4], 0, 0);
  }
  wait_async_all();
#else
  float4* o4 = (float4*)obase;
  for (int idx = tid; idx < MCH * RWS * HH / 4; idx += 256) {
    int mm  = idx >> 7;
    int rem = idx & 127;
    float4 vv = *(const float4*)&s_out[mm * SLC + rem * 4];
    o4[(size_t)mm * (HH * HH / 4) + rem] = vv;
  }
#endif
}

extern "C" void kernel_launch(void* const* d_in, const int* in_sizes, int n_in,
                              void* d_out, int out_size, void* d_ws, size_t ws_size,
                              hipStream_t stream) {
  const float* in   = (const float*)d_in[0];  // 128*1*128*128
  const float* wgt  = (const float*)d_in[1];  // 1024*9
  const float* bias = (const float*)d_in[2];  // 1024
  float* out = (float*)d_out;                 // 128*16*128*128
  (void)in_sizes; (void)n_in; (void)out_size; (void)d_ws; (void)ws_size;
  dim3 grid(NB * (HH / RWS));   // 128 images * 32 row-chunks = 4096 blocks
  shiftvar_wmma_kernel<<<grid, dim3(256), 0, stream>>>(in, wgt, bias, out);
}